// CrossPointNet_91070486544467
// MI455X (gfx1250) — compile-verified
//
#include <hip/hip_runtime.h>
#include <hip/hip_bf16.h>
#include <stdint.h>

typedef __attribute__((ext_vector_type(2))) float v2f;
typedef __attribute__((ext_vector_type(8))) float v8f;
typedef __attribute__((ext_vector_type(4))) int v4i;

#if __has_builtin(__builtin_amdgcn_global_load_async_to_lds_b128)
#define CPN_HAS_ASYNC_LDS 1
typedef __attribute__((address_space(1))) v4i* cpn_gptr;   // global int4*
typedef __attribute__((address_space(3))) v4i* cpn_lptr;   // LDS int4*
#endif

namespace {
constexpr int kB    = 4;
constexpr int kN    = 16384;
constexpr int kS    = 64;
constexpr int kFill = 6144;   // NUM_FILL
constexpr int kCut  = 256;    // CUT
constexpr int kNPS  = 64;     // N_PER_SAMPLE
constexpr int kPerTh = kFill / 256;  // 24 strided elements per thread

// workspace layout (bytes) — total ~10.6 MB, L2-resident on MI455X
constexpr size_t WS_RADIUS = 0;                                   // 256 f32
constexpr size_t WS_CNORM  = 1024;                                // 256 f32
constexpr size_t WS_SCOORD = 2048;                                // 256 float4
constexpr size_t WS_PNORM  = 8192;                                // 65536 f32
constexpr size_t WS_MASK   = 512 * 1024;                          // B*S*N u8 = 4 MiB
constexpr size_t WS_NEIGH  = WS_MASK + (size_t)kB * kS * kN;      // B*S*kFill i32 = 6 MiB
}

// ---------------------------------------------------------------------------
// Stage 1: gather sampled centers, nearest-other-center radius², center norms.
// ---------------------------------------------------------------------------
__global__ void cpn_centers(const float4* __restrict__ pts, const int* __restrict__ centers,
                            float* __restrict__ out_coords, float* __restrict__ radius,
                            float* __restrict__ cnorm, float4* __restrict__ scoord) {
  __shared__ float sx[kS], sy[kS], sz[kS];
  const int b = blockIdx.x, s = threadIdx.x;
  int ci = centers[b * kS + s];
  ci = ci < 0 ? 0 : (ci >= kN ? kN - 1 : ci);
  const float4 p = pts[(size_t)b * kN + ci];
  out_coords[(b * kS + s) * 3 + 0] = p.x;
  out_coords[(b * kS + s) * 3 + 1] = p.y;
  out_coords[(b * kS + s) * 3 + 2] = p.z;
  scoord[b * kS + s] = make_float4(p.x, p.y, p.z, 0.0f);
  cnorm[b * kS + s] = p.x * p.x + p.y * p.y + p.z * p.z;
  sx[s] = p.x; sy[s] = p.y; sz[s] = p.z;
  __syncthreads();
  float mn = 3.0e38f;
  for (int t = 0; t < kS; ++t) {
    if (t == s) continue;
    const float dx = sx[t] - p.x, dy = sy[t] - p.y, dz = sz[t] - p.z;
    mn = fminf(mn, dx * dx + dy * dy + dz * dz);
  }
  radius[b * kS + s] = mn * 2.25f;   // * 1.5^2
}

// ---------------------------------------------------------------------------
// Stage 2a: per-point squared norms (for ||p-c||^2 = pn + cn - 2 p.c).
// ---------------------------------------------------------------------------
__global__ void cpn_pnorm(const float4* __restrict__ pts, float* __restrict__ pnorm) {
  const int i = blockIdx.x * blockDim.x + threadIdx.x;   // 0..B*N-1
  const float4 p = pts[i];
  pnorm[i] = p.x * p.x + p.y * p.y + p.z * p.z;
}

// ---------------------------------------------------------------------------
// Stage 2b: within-radius mask via V_WMMA_F32_16X16X4_F32.
// One wave = one 16-point A tile x ALL four 16-center B tiles: the A operand
// and the 8 point norms are loaded once and reused across 4 WMMAs.
// 8 waves / 256-thread block; 4096 wave-tasks → EXEC all-ones everywhere.
// ---------------------------------------------------------------------------
__global__ __launch_bounds__(256) void cpn_mask(const float4* __restrict__ pts,
                                                const float4* __restrict__ scoord,
                                                const float* __restrict__ pnorm,
                                                const float* __restrict__ cnorm,
                                                const float* __restrict__ radius,
                                                uint8_t* __restrict__ mask) {
  const int wave  = threadIdx.x >> 5;
  const int lane  = threadIdx.x & 31;
  const int task  = blockIdx.x * 8 + wave;   // b(2) | ptile(10)
  const int ptile = task & 1023;
  const int b     = task >> 10;
  const int col   = lane & 15;               // A row M, B col N, D col N
  const int half  = lane >> 4;               // K half: {0,1}->(x,y), {2,3}->(z,0)

  // A operand: point row M = col, VGPR0/VGPR1 = K{0,2}/{1,3} by lane half
  const float4 P = pts[(size_t)b * kN + ptile * 16 + col];
  v2f a;
  a.x = half ? P.z : P.x;
  a.y = half ? 0.0f : P.y;

  // point norms for the 8 D rows owned by this lane (M = rbase..rbase+7)
  const int rbase = ptile * 16 + half * 8;
  const float4* pn4 = (const float4*)(pnorm + (size_t)b * kN + rbase);
  const float4 pnA = pn4[0], pnB = pn4[1];
  const float pn[8] = {pnA.x, pnA.y, pnA.z, pnA.w, pnB.x, pnB.y, pnB.z, pnB.w};

#pragma unroll
  for (int ct = 0; ct < 4; ++ct) {
    // B operand: center col N = col, same K striping as A
    const int c = ct * 16 + col;
    const float4 C = scoord[b * kS + c];
    v2f bm;
    bm.x = half ? C.z : C.x;
    bm.y = half ? 0.0f : C.y;

    v8f acc = {};
    acc = __builtin_amdgcn_wmma_f32_16x16x4_f32(
        /*neg_a=*/false, a, /*neg_b=*/false, bm,
        /*c_mod=*/(short)0, acc, /*reuse_a=*/false, /*reuse_b=*/false);

    const float cn  = cnorm[b * kS + c];
    const float rad = radius[b * kS + c];
    uint32_t m0 = 0, m1 = 0;
#pragma unroll
    for (int r = 0; r < 8; ++r) {
      const float dist = pn[r] + cn - 2.0f * acc[r];
      const uint32_t w = (dist <= rad) ? 1u : 0u;
      if (r < 4) m0 |= w << (8 * r); else m1 |= w << (8 * (r - 4));
    }
    // 8 contiguous mask bytes for this (center, point-octet): one 8-byte store
    uint2* dst = (uint2*)(mask + (size_t)(b * kS + c) * kN + rbase);
    dst[0] = make_uint2(m0, m1);
  }
}

// ---------------------------------------------------------------------------
// Stage 3: _process_mask — ordered compaction of first kFill set indices,
// pad the tail with the center index. One block per (b,s); wave32 ballots.
// ---------------------------------------------------------------------------
__global__ __launch_bounds__(256) void cpn_compact(const uint8_t* __restrict__ mask,
                                                   const int* __restrict__ centers,
                                                   int* __restrict__ neigh) {
  const int g = blockIdx.x;
  const int b = g >> 6, s = g & 63;
  const int tid = threadIdx.x;
  const int lane = tid & 31, wave = tid >> 5;
  __shared__ int s_wcnt[8];
  __shared__ int s_run;
  if (tid == 0) s_run = 0;
  __syncthreads();

  const uint8_t* mrow = mask + (size_t)g * kN;
  int* nrow = neigh + (size_t)g * kFill;

  for (int base = 0; base < kN; base += 256) {
    if (base + 256 < kN) __builtin_prefetch(mrow + base + 256 + tid, 0, 1);
    const int n = base + tid;
    const int m = mrow[n];
    const uint32_t bal = (uint32_t)__ballot(m != 0);
    const int rank = __popc(bal & ((1u << lane) - 1u));
    if (lane == 0) s_wcnt[wave] = __popc(bal);
    __syncthreads();
    int woff = 0;
    for (int w = 0; w < wave; ++w) woff += s_wcnt[w];
    const int pos = s_run + woff + rank;
    if (m && pos < kFill) nrow[pos] = n;
    __syncthreads();
    if (tid == 0) {
      int t = 0;
      for (int w = 0; w < 8; ++w) t += s_wcnt[w];
      s_run += t;
    }
    __syncthreads();
    if (s_run >= kFill) break;   // uniform
  }

  int count = s_run < kFill ? s_run : kFill;
  if (count < 1) count = 1;      // matches max(counts, 1)
  int ci = centers[b * kS + s];
  ci = ci < 0 ? 0 : (ci >= kN ? kN - 1 : ci);
  for (int pos = count + tid; pos < kFill; pos += 256) nrow[pos] = ci;
}

// ---------------------------------------------------------------------------
// Stage 4: FPS, 64 steps per group. Neighbor indices are staged into LDS once
// via GLOBAL_LOAD_ASYNC_TO_LDS_B128 (ASYNCcnt path) and hit LDS for all 64
// steps. Per-step distance keys + running min live in registers (each thread
// owns exactly 24 strided elements). top-256 = binary-search k-th selection on
// 45-bit keys (distbits<<13 | idx, unique → exact & stable), ordered ballot
// compaction, then a 256-wide bitonic sort → jax.top_k(-dist) order exactly.
// LDS ≈ 29 KB / workgroup of the 320 KB WGP pool.
// ---------------------------------------------------------------------------
__global__ __launch_bounds__(256) void cpn_fps(const float4* __restrict__ pts,
                                               const int* __restrict__ neigh,
                                               float* __restrict__ out_result,
                                               float* __restrict__ out_local) {
  const int g = blockIdx.x;
  const int b = g >> 6;
  const int tid = threadIdx.x;
  const int lane = tid & 31, wave = tid >> 5;

  __shared__ int s_nidx[kFill];                   // 24 KB neighbor indices
  __shared__ float s_rv[256];
  __shared__ int s_ri[256];
  __shared__ unsigned long long s_sel[kCut];      // selected (key<<13|idx)
  __shared__ int c_wcnt[8];
  __shared__ int c_run;
  __shared__ int s_far;

  const int* nrow = neigh + (size_t)g * kFill;
  const float4* bpts = pts + (size_t)b * kN;

  // ---- stage neighbor row into LDS (async copy, tracked by ASYNCcnt) ----
#ifdef CPN_HAS_ASYNC_LDS
#pragma unroll
  for (int j = 0; j < kFill * 4 / (256 * 16); ++j) {   // 6 x 4 KB
    const int off = (j * 256 + tid) * 16;
    __builtin_amdgcn_global_load_async_to_lds_b128(
        (cpn_gptr)((const char*)nrow + off),
        (cpn_lptr)((char*)s_nidx + off),
        0, 0);
  }
#if __has_builtin(__builtin_amdgcn_s_wait_asynccnt)
  __builtin_amdgcn_s_wait_asynccnt(0);
#else
  asm volatile("s_wait_asynccnt 0" ::: "memory");
#endif
#else
  for (int i = tid; i < kFill; i += 256) s_nidx[i] = nrow[i];
#endif
  if (tid == 0) s_far = 0;

  float mreg[kPerTh];       // running min distance, element i = j*256 + tid
  uint32_t kreg[kPerTh];    // current-step distance bits
#pragma unroll
  for (int j = 0; j < kPerTh; ++j) mreg[j] = 1.0e10f;
  __syncthreads();

  for (int step = 0; step < kNPS; ++step) {
    const int far = s_far;
    const int farN = s_nidx[far];
    if (tid == 0) out_result[(size_t)g * kNPS + step] = (float)farN;
    const float4 cp = bpts[farN];

#pragma unroll
    for (int j = 0; j < kPerTh; ++j) {
      const int i = j * 256 + tid;
      const float4 p = bpts[s_nidx[i]];
      const float dx = p.x - cp.x, dy = p.y - cp.y, dz = p.z - cp.z;
      const float d = dx * dx + dy * dy + dz * dz;
      kreg[j] = __float_as_uint(d);      // d >= 0 → bits order = value order
      mreg[j] = fminf(mreg[j], d);
    }

    // ---- k-th (256th) smallest combined key, binary search over 45 bits ----
    unsigned long long lo = 0ull;
    unsigned long long hi = (0xFFFFFFFFull << 13) | 0x1FFFull;
    while (lo < hi) {
      const unsigned long long mid = lo + ((hi - lo) >> 1);
      int c = 0;
#pragma unroll
      for (int j = 0; j < kPerTh; ++j) {
        const unsigned long long K =
            ((unsigned long long)kreg[j] << 13) | (unsigned)(j * 256 + tid);
        c += (K <= mid) ? 1 : 0;
      }
      s_ri[tid] = c;
      __syncthreads();
      for (int off = 128; off > 0; off >>= 1) {
        if (tid < off) s_ri[tid] += s_ri[tid + off];
        __syncthreads();
      }
      const int total = s_ri[0];
      __syncthreads();
      if (total >= kCut) hi = mid; else lo = mid + 1;
    }
    const unsigned long long kth = lo;             // exactly kCut keys <= kth

    // ---- ordered compaction of the kCut selected keys ----
    s_sel[tid] = ~0ull;
    if (tid == 0) c_run = 0;
    __syncthreads();
#pragma unroll
    for (int j = 0; j < kPerTh; ++j) {
      const unsigned long long K =
          ((unsigned long long)kreg[j] << 13) | (unsigned)(j * 256 + tid);
      const int sel = (K <= kth) ? 1 : 0;
      const uint32_t bal = (uint32_t)__ballot(sel);
      const int rank = __popc(bal & ((1u << lane) - 1u));
      if (lane == 0) c_wcnt[wave] = __popc(bal);
      __syncthreads();
      int woff = 0;
      for (int w = 0; w < wave; ++w) woff += c_wcnt[w];
      const int pos = c_run + woff + rank;
      if (sel && pos < kCut) s_sel[pos] = K;
      __syncthreads();
      if (tid == 0) {
        int t = 0;
        for (int w = 0; w < 8; ++w) t += c_wcnt[w];
        c_run += t;
      }
      __syncthreads();
    }

    // ---- bitonic sort ascending by (dist, idx) ----
    for (int k = 2; k <= kCut; k <<= 1) {
      for (int j = k >> 1; j > 0; j >>= 1) {
        const int ixj = tid ^ j;
        if (ixj > tid) {
          const bool up = ((tid & k) == 0);
          const unsigned long long A = s_sel[tid];
          const unsigned long long Bv = s_sel[ixj];
          if ((A > Bv) == up) { s_sel[tid] = Bv; s_sel[ixj] = A; }
        }
        __syncthreads();
      }
    }

    // ---- local_idx: map local ranks back to original point indices ----
    {
      const int li = (int)(s_sel[tid] & 0x1FFFull);
      out_local[((size_t)g * kNPS + step) * kCut + tid] = (float)s_nidx[li];
    }

    // ---- argmax of running min (first occurrence wins ties) ----
    float bv = -1.0f;
    int bi = 0;
#pragma unroll
    for (int j = 0; j < kPerTh; ++j) {
      if (mreg[j] > bv) { bv = mreg[j]; bi = j * 256 + tid; }
    }
    s_rv[tid] = bv; s_ri[tid] = bi;
    __syncthreads();
    for (int off = 128; off > 0; off >>= 1) {
      if (tid < off) {
        const float ov = s_rv[tid + off];
        const int oi = s_ri[tid + off];
        if (ov > s_rv[tid] || (ov == s_rv[tid] && oi < s_ri[tid])) {
          s_rv[tid] = ov; s_ri[tid] = oi;
        }
      }
      __syncthreads();
    }
    if (tid == 0) s_far = s_ri[0];
    __syncthreads();
  }
}

// ---------------------------------------------------------------------------
extern "C" void kernel_launch(void* const* d_in, const int* in_sizes, int n_in,
                              void* d_out, int out_size, void* d_ws, size_t ws_size,
                              hipStream_t stream) {
  (void)in_sizes; (void)n_in; (void)out_size; (void)ws_size;
  const float4* pts = (const float4*)d_in[0];      // (4,16384,4) f32
  const int* centers = (const int*)d_in[1];        // (4,64) i32
  float* out = (float*)d_out;
  char* ws = (char*)d_ws;

  float*   radius = (float*)(ws + WS_RADIUS);
  float*   cnorm  = (float*)(ws + WS_CNORM);
  float4*  scoord = (float4*)(ws + WS_SCOORD);
  float*   pnorm  = (float*)(ws + WS_PNORM);
  uint8_t* mask   = (uint8_t*)(ws + WS_MASK);
  int*     neigh  = (int*)(ws + WS_NEIGH);

  float* out_coords = out;                                  // 4*64*3   = 768
  float* out_result = out + 768;                            // 4*64*64  = 16384
  float* out_local  = out + 768 + kB * kS * kNPS;           // 4*64*64*256

  cpn_centers<<<kB, kS, 0, stream>>>(pts, centers, out_coords, radius, cnorm, scoord);
  cpn_pnorm<<<(kB * kN) / 256, 256, 0, stream>>>(pts, pnorm);
  cpn_mask<<<(kB * 1024) / 8, 256, 0, stream>>>(pts, scoord, pnorm, cnorm, radius, mask);
  cpn_compact<<<kB * kS, 256, 0, stream>>>(mask, centers, neigh);
  cpn_fps<<<kB * kS, 256, 0, stream>>>(pts, neigh, out_result, out_local);
}